// SequenceConvolution_81466939670707
// MI455X (gfx1250) — compile-verified
//
#include <hip/hip_runtime.h>
#include <math.h>

typedef float v2f __attribute__((ext_vector_type(2)));
typedef float v8f __attribute__((ext_vector_type(8)));

#define SEQ_DIM 320
#define C0 128
#define C1 64
#define KW 5
#define EPSF 1e-6f
#define SQRT3F 1.7320508075688772f
#define INV_SQRT_KS 0.03952847075210474f   /* 1/sqrt(5*128) */
#define INV_SQRT_KV 0.05383819020581655f   /* 1/sqrt(5*64+25) */

#define WAVES_PER_BLOCK 2
#define ROWS_PER_WAVE 16
#define ROWS_PER_BLOCK (WAVES_PER_BLOCK * ROWS_PER_WAVE)
#define LDS_SH_FLOATS 1200      /* 16 rows * 25 pairs * 3 */
#define LDS_BUF_FLOATS 3072     /* max(16*128, 16*64*3)   */
#define LDS_PER_WAVE (LDS_SH_FLOATS + LDS_BUF_FLOATS)

__global__ __launch_bounds__(WAVES_PER_BLOCK * 32)
void seqconv_kernel(const float* __restrict__ x,
                    const float* __restrict__ coord,
                    const int*   __restrict__ mask_ia,
                    const int*   __restrict__ mask_c,
                    const float* __restrict__ Ws,   /* [640,128] */
                    const float* __restrict__ Wv,   /* [345, 64] */
                    float* __restrict__ out,
                    int n)
{
  __shared__ float lds[WAVES_PER_BLOCK * LDS_PER_WAVE];
  const int tid   = threadIdx.x;
  const int wave  = tid >> 5;
  const int lane  = tid & 31;
  const int laneM = lane & 15;
  const int hi    = lane >> 4;           /* 0: K={0,1}, 1: K={2,3} */

  float* shW  = lds + wave * LDS_PER_WAVE;   /* sh tile: [16][25][3] */
  float* bufW = shW + LDS_SH_FLOATS;         /* OS / OV staging      */

  const int rowbase = (blockIdx.x * WAVES_PER_BLOCK + wave) * ROWS_PER_WAVE;
  const int myrow   = rowbase + laneM;
  const bool rowok  = myrow < n;
  const int row     = rowok ? myrow : (n - 1);

  /* ---------------- Phase A: masks, row pointers, SH pairs ---------------- */
  const float* xp[KW];
  float mIA[KW], mC[KW];
  float gcx[KW], gcy[KW], gcz[KW];
  float sIA = 0.f, sC = 0.f;
#pragma unroll
  for (int j = 0; j < KW; ++j) {
    int src = row + j - 2;
    bool valid = (src >= 0) && (src < n);
    int safe = valid ? src : 0;
    float ia = (valid && (mask_ia[safe] != 0)) ? 1.f : 0.f;
    float mc = (valid && (mask_c[safe]  != 0)) ? 1.f : 0.f;
    xp[j] = x + (size_t)safe * SEQ_DIM;
    mIA[j] = ia; mC[j] = mc;
    sIA += ia; sC += mc;
    gcx[j] = coord[safe * 3 + 0] * mc;
    gcy[j] = coord[safe * 3 + 1] * mc;
    gcz[j] = coord[safe * 3 + 2] * mc;
  }
  {
    float* shrow = shW + laneM * 75;
#pragma unroll
    for (int p = 0; p < 25; ++p) {
      const int j1 = p / 5, j2 = p % 5;
      if ((p < 13) == (hi == 0)) {       /* split the 25 pairs across lane halves */
        float pm = mC[j1] * mC[j2];
        float dx = (gcx[j1] - gcx[j2]) * pm;
        float dy = (gcy[j1] - gcy[j2]) * pm;
        float dz = (gcz[j1] - gcz[j2]) * pm;
        float r2 = dx * dx + dy * dy + dz * dz;
        float rin = (r2 > 0.f) ? (SQRT3F / sqrtf(r2)) : 0.f;
        shrow[p * 3 + 0] = dx * rin;
        shrow[p * 3 + 1] = dy * rin;
        shrow[p * 3 + 2] = dz * rin;
      }
    }
  }
  if (hi == 0 && rowok) {
    out[(size_t)n * 320 + (size_t)row * 3 + 0] = coord[row * 3 + 0];
    out[(size_t)n * 320 + (size_t)row * 3 + 1] = coord[row * 3 + 1];
    out[(size_t)n * 320 + (size_t)row * 3 + 2] = coord[row * 3 + 2];
    out[(size_t)n * 323 + row] = (sIA > 0.5f) ? 1.f : 0.f;
    out[(size_t)n * 324 + row] = (sC  > 0.5f) ? 1.f : 0.f;
  }
  __syncthreads();

  const v8f z8 = {0.f, 0.f, 0.f, 0.f, 0.f, 0.f, 0.f, 0.f};

  /* ---------------- Phase B: scalar GEMM [16x640]x[640x128] -------------- */
  {
    v8f acc[8];
#pragma unroll
    for (int t = 0; t < 8; ++t) acc[t] = z8;
#pragma unroll
    for (int j = 0; j < KW; ++j) {
      const float* xr = xp[j];
      const float mk = mIA[j];
      for (int kk = 0; kk < 32; ++kk) {
        const int o = 4 * kk + 2 * hi;       /* 0..126, even */
        v2f A;
        A.x = xr[o] * mk;
        A.y = xr[o + 1] * mk;
        const int m0 = j * 128 + o;
#pragma unroll
        for (int t = 0; t < 8; ++t) {
          const int col = t * 16 + laneM;
          v2f B;
          B.x = Ws[(size_t)m0 * 128 + col];
          B.y = Ws[(size_t)m0 * 128 + 128 + col];
          acc[t] = __builtin_amdgcn_wmma_f32_16x16x4_f32(
              false, A, false, B, (short)0, acc[t], false, false);
        }
      }
    }
    /* stage OS tile into LDS: [row][ch] */
#pragma unroll
    for (int t = 0; t < 8; ++t)
#pragma unroll
      for (int v = 0; v < 8; ++v)
        bufW[(v + 8 * hi) * 128 + t * 16 + laneM] = acc[t][v] * INV_SQRT_KS;
  }
  __syncthreads();

  /* LayerNorm over 128 scalar channels, then store */
  {
    const float* rp = bufW + laneM * 128 + hi * 64;
    float s = 0.f;
    for (int c = 0; c < 64; ++c) s += rp[c];
    s += __shfl_xor(s, 16, 32);
    const float mu = s * (1.f / 128.f);
    float s2 = 0.f;
    for (int c = 0; c < 64; ++c) { float d = rp[c] - mu; s2 += d * d; }
    s2 += __shfl_xor(s2, 16, 32);
    const float rin = 1.f / sqrtf(s2 * (1.f / 128.f) + EPSF);
    if (rowok) {
      float* op = out + (size_t)row * 320 + hi * 64;
      for (int c = 0; c < 64; ++c) op[c] = (rp[c] - mu) * rin;
    }
  }
  __syncthreads();

  /* ------------- Phase C: vector GEMM 3 x [16x348]x[348x64] -------------- */
  {
    v8f av[3][4];
#pragma unroll
    for (int i = 0; i < 3; ++i)
#pragma unroll
      for (int t = 0; t < 4; ++t) av[i][t] = z8;

#pragma unroll
    for (int j = 0; j < KW; ++j) {
      const float* xr = xp[j] + C0;
      const float mk = mIA[j];
      for (int kk = 0; kk < 16; ++kk) {
        const int o = 4 * kk + 2 * hi;       /* channel base 0..62 */
        const int m0 = j * 64 + o;
        float a0x = xr[3 * o + 0] * mk, a0y = xr[3 * o + 1] * mk, a0z = xr[3 * o + 2] * mk;
        float a1x = xr[3 * o + 3] * mk, a1y = xr[3 * o + 4] * mk, a1z = xr[3 * o + 5] * mk;
#pragma unroll
        for (int t = 0; t < 4; ++t) {
          const int col = t * 16 + laneM;
          v2f B;
          B.x = Wv[(size_t)m0 * 64 + col];
          B.y = Wv[(size_t)m0 * 64 + 64 + col];
          v2f A0; A0.x = a0x; A0.y = a1x;
          av[0][t] = __builtin_amdgcn_wmma_f32_16x16x4_f32(
              false, A0, false, B, (short)0, av[0][t], false, false);
          v2f A1; A1.x = a0y; A1.y = a1y;
          av[1][t] = __builtin_amdgcn_wmma_f32_16x16x4_f32(
              false, A1, false, B, (short)0, av[1][t], false, false);
          v2f A2; A2.x = a0z; A2.y = a1z;
          av[2][t] = __builtin_amdgcn_wmma_f32_16x16x4_f32(
              false, A2, false, B, (short)0, av[2][t], false, false);
        }
      }
    }
    /* SH tail: K = 320..347 (padded to 348) */
    const float* shrow = shW + laneM * 75;
#pragma unroll
    for (int kk = 0; kk < 7; ++kk) {
      const int m0 = 320 + 4 * kk + 2 * hi;
      const int p0 = m0 - 320;
      const int p1 = p0 + 1;
      float a0x = 0.f, a0y = 0.f, a0z = 0.f, a1x = 0.f, a1y = 0.f, a1z = 0.f;
      if (p0 < 25) { a0x = shrow[p0 * 3 + 0]; a0y = shrow[p0 * 3 + 1]; a0z = shrow[p0 * 3 + 2]; }
      if (p1 < 25) { a1x = shrow[p1 * 3 + 0]; a1y = shrow[p1 * 3 + 1]; a1z = shrow[p1 * 3 + 2]; }
#pragma unroll
      for (int t = 0; t < 4; ++t) {
        const int col = t * 16 + laneM;
        v2f B;
        B.x = (m0 < 345)     ? Wv[(size_t)m0 * 64 + col]      : 0.f;
        B.y = (m0 + 1 < 345) ? Wv[(size_t)m0 * 64 + 64 + col] : 0.f;
        v2f A0; A0.x = a0x; A0.y = a1x;
        av[0][t] = __builtin_amdgcn_wmma_f32_16x16x4_f32(
            false, A0, false, B, (short)0, av[0][t], false, false);
        v2f A1; A1.x = a0y; A1.y = a1y;
        av[1][t] = __builtin_amdgcn_wmma_f32_16x16x4_f32(
            false, A1, false, B, (short)0, av[1][t], false, false);
        v2f A2; A2.x = a0z; A2.y = a1z;
        av[2][t] = __builtin_amdgcn_wmma_f32_16x16x4_f32(
            false, A2, false, B, (short)0, av[2][t], false, false);
      }
    }
    /* stage OV tile into LDS: [row][ch][xyz] */
#pragma unroll
    for (int t = 0; t < 4; ++t)
#pragma unroll
      for (int v = 0; v < 8; ++v) {
        float* bp = bufW + ((v + 8 * hi) * 64 + t * 16 + laneM) * 3;
        bp[0] = av[0][t][v] * INV_SQRT_KV;
        bp[1] = av[1][t][v] * INV_SQRT_KV;
        bp[2] = av[2][t][v] * INV_SQRT_KV;
      }
  }
  __syncthreads();

  /* Vector RMS norm over 64 channel norms, then store */
  {
    const float* rp = bufW + (laneM * 64 + hi * 32) * 3;
    float s = 0.f;
    for (int c = 0; c < 32; ++c) {
      float vx = rp[c * 3 + 0], vy = rp[c * 3 + 1], vz = rp[c * 3 + 2];
      s += vx * vx + vy * vy + vz * vz;
    }
    s += __shfl_xor(s, 16, 32);
    const float rin = 1.f / sqrtf(s * (1.f / 64.f) + EPSF);
    if (rowok) {
      float* op = out + (size_t)row * 320 + C0 + hi * 96;
      for (int c = 0; c < 32; ++c) {
        op[c * 3 + 0] = rp[c * 3 + 0] * rin;
        op[c * 3 + 1] = rp[c * 3 + 1] * rin;
        op[c * 3 + 2] = rp[c * 3 + 2] * rin;
      }
    }
  }
}

extern "C" void kernel_launch(void* const* d_in, const int* in_sizes, int n_in,
                              void* d_out, int out_size, void* d_ws, size_t ws_size,
                              hipStream_t stream) {
  (void)n_in; (void)out_size; (void)d_ws; (void)ws_size;
  const float* x     = (const float*)d_in[0];
  const float* coord = (const float*)d_in[1];
  const int*   m_ia  = (const int*)d_in[2];
  const int*   m_c   = (const int*)d_in[3];
  const float* Ws    = (const float*)d_in[4];
  const float* Wv    = (const float*)d_in[5];
  float* out = (float*)d_out;
  const int n = in_sizes[0] / SEQ_DIM;
  const int blocks = (n + ROWS_PER_BLOCK - 1) / ROWS_PER_BLOCK;
  seqconv_kernel<<<blocks, WAVES_PER_BLOCK * 32, 0, stream>>>(
      x, coord, m_ia, m_c, Ws, Wv, out, n);
}